// PhaseLinear_58463094833618
// MI455X (gfx1250) — compile-verified
//
#include <hip/hip_runtime.h>
#include <hip/hip_bf16.h>

// PhaseLinear fused as a single K=2048 f32 WMMA GEMM:
//   out[b,o] = sum_c sum_i (coef[b,c]*input[b,i]) * W[c,o,i] + sum_c coef[b,c]*bias[c,o]
// The f32 WMMA A-fragment keeps each lane on a single A-row (M = lane&15), so the
// per-row spline coefficient is applied as a register-side fragment scale. The raw
// input tile is staged to LDS once per K-chunk and reused across all 4 control
// points, whose B tiles are co-resident in LDS.

constexpr int BATCH = 1024;
constexpr int IN_F  = 512;
constexpr int OUT_F = 512;
constexpr int NCP   = 4;

constexpr int BM = 64;             // block tile rows (batch)
constexpr int BN = 64;             // block tile cols (out features)
constexpr int BK = 32;             // K chunk staged in LDS
constexpr int LDS_STRIDE = BK + 4; // 36 floats: rows 144B (16B aligned) + conflict-free frag reads
constexpr int THREADS = 128;       // 4 wave32 waves

typedef __attribute__((ext_vector_type(2))) float v2f;
typedef __attribute__((ext_vector_type(8))) float v8f;

__device__ __forceinline__ float spline_t(float ph) {
    constexpr float PI = 3.14159265358979323846f;
    float t0 = ph * (1.0f / (1.5f * PI));
    float t1 = (ph - 0.5f * PI) * (1.0f / (1.5f * PI));
    return (ph < 1.5f * PI) ? t0 : t1;
}

// Faithful to the reference (tt = [1, t^2, t^3, 0] @ Catmull-Rom basis):
// cf = [t^3 - 0.5 t^2, 1 - 2.5 t^3, 0.5 t^2 + 2 t^3, -0.5 t^3]
__device__ __forceinline__ void spline_coefs(float ph, float cf[4]) {
    float t  = spline_t(ph);
    float t2 = t * t;
    float t3 = t2 * t;
    cf[0] = t3 - 0.5f * t2;
    cf[1] = 1.0f - 2.5f * t3;
    cf[2] = 0.5f * t2 + 2.0f * t3;
    cf[3] = -0.5f * t3;
}

__global__ __launch_bounds__(THREADS)
void phase_linear_wmma_kernel(const float* __restrict__ input,
                              const float* __restrict__ phase,
                              const float* __restrict__ weights,
                              const float* __restrict__ biases,
                              float* __restrict__ out) {
    __shared__ __align__(16) float ldsA[BM * LDS_STRIDE];               //  9 KB
    __shared__ __align__(16) float ldsB[NCP * BN * LDS_STRIDE];         // 36 KB

    const int tid    = threadIdx.x;
    const int lane   = tid & 31;
    const int wave   = tid >> 5;       // 0..3
    const int mw     = wave >> 1;      // 0..1 : 32-row wave tile
    const int nw     = wave & 1;       // 0..1 : 32-col wave tile
    const int lane15 = lane & 15;
    const int laneHi = lane >> 4;      // 0 or 1 (selects K+2 / row+8 half)

    const int row0 = blockIdx.y * BM;
    const int col0 = blockIdx.x * BN;

    // Per-lane A-row spline coefficients: the f32 WMMA A-fragment holds only
    // row M = lane&15, so each lane needs the 4 coefs of exactly 2 rows.
    float cfr[2][4];
#pragma unroll
    for (int mt = 0; mt < 2; ++mt)
        spline_coefs(phase[row0 + mw * 32 + mt * 16 + lane15], cfr[mt]);

    const v8f vzero = {0.f, 0.f, 0.f, 0.f, 0.f, 0.f, 0.f, 0.f};
    v8f acc[2][2];
#pragma unroll
    for (int mt = 0; mt < 2; ++mt)
#pragma unroll
        for (int nt = 0; nt < 2; ++nt)
            acc[mt][nt] = vzero;

    for (int k0 = 0; k0 < IN_F; k0 += BK) {
        __syncthreads();
        // ---- Stage raw A tile (input), reused by all 4 control points
#pragma unroll
        for (int t = 0; t < (BM * BK) / (4 * THREADS); ++t) {          // 4 iters
            int idx = tid + t * THREADS;
            int r   = idx >> 3;            // 0..63
            int c4  = (idx & 7) << 2;      // 0,4,...,28
            float4 v = *(const float4*)(input + (row0 + r) * IN_F + k0 + c4);
            *(float4*)(ldsA + r * LDS_STRIDE + c4) = v;
        }
        // ---- Stage B tiles for all 4 control points: weights[c][col][k]
#pragma unroll
        for (int t = 0; t < (NCP * BN * BK) / (4 * THREADS); ++t) {    // 16 iters
            int idx = tid + t * THREADS;   // 0..2047
            int c   = idx >> 9;            // 0..3
            int rem = idx & 511;
            int n   = rem >> 3;            // 0..63
            int c4  = (rem & 7) << 2;      // 0,4,...,28
            float4 v = *(const float4*)(weights +
                        ((size_t)c * OUT_F + col0 + n) * IN_F + k0 + c4);
            *(float4*)(ldsB + (c * BN + n) * LDS_STRIDE + c4) = v;
        }
        // ---- Prefetch next K chunk into cache (global_prefetch_b8); k0 uniform
        if (k0 + BK < IN_F) {
            __builtin_prefetch(input + (row0 + (tid >> 1)) * IN_F + k0 + BK, 0, 1);
            __builtin_prefetch(weights + ((size_t)(tid >> 5) * OUT_F + col0 +
                               ((tid & 31) << 1)) * IN_F + k0 + BK, 0, 1);
        }
        __syncthreads();

        // ---- 8 WMMA K-steps of 4; A fragment shared across control points
#pragma unroll
        for (int kk = 0; kk < BK; kk += 4) {
            const int kb = kk + (laneHi << 1);   // lanes 16-31 hold K+2,K+3
            v2f a[2];
#pragma unroll
            for (int mt = 0; mt < 2; ++mt) {
                const float* p = ldsA + (mw * 32 + mt * 16 + lane15) * LDS_STRIDE + kb;
                a[mt][0] = p[0];
                a[mt][1] = p[1];
            }
#pragma unroll
            for (int c = 0; c < NCP; ++c) {
                v2f as[2], b[2];
#pragma unroll
                for (int mt = 0; mt < 2; ++mt)
                    as[mt] = a[mt] * cfr[mt][c];   // per-row coef, register side
#pragma unroll
                for (int nt = 0; nt < 2; ++nt) {
                    const float* p = ldsB + (c * BN + nw * 32 + nt * 16 + lane15)
                                     * LDS_STRIDE + kb;
                    b[nt][0] = p[0];
                    b[nt][1] = p[1];
                }
#pragma unroll
                for (int mt = 0; mt < 2; ++mt)
#pragma unroll
                    for (int nt = 0; nt < 2; ++nt)
                        acc[mt][nt] = __builtin_amdgcn_wmma_f32_16x16x4_f32(
                            false, as[mt], false, b[nt],
                            (short)0, acc[mt][nt], false, false);
            }
        }
    }

    // ---- Epilogue: add spline-blended bias, write out.
    // C/D layout: VGPR j -> row j (+8 for lanes 16-31), col = lane&15.
#pragma unroll
    for (int mt = 0; mt < 2; ++mt) {
#pragma unroll
        for (int nt = 0; nt < 2; ++nt) {
            const int col = col0 + nw * 32 + nt * 16 + lane15;
#pragma unroll
            for (int j = 0; j < 8; ++j) {
                const int r = row0 + mw * 32 + mt * 16 + (laneHi << 3) + j;
                float cf[4];
                spline_coefs(phase[r], cf);
                float bias = cf[0] * biases[col]
                           + cf[1] * biases[OUT_F + col]
                           + cf[2] * biases[2 * OUT_F + col]
                           + cf[3] * biases[3 * OUT_F + col];
                out[r * OUT_F + col] = acc[mt][nt][j] + bias;
            }
        }
    }
}

extern "C" void kernel_launch(void* const* d_in, const int* in_sizes, int n_in,
                              void* d_out, int out_size, void* d_ws, size_t ws_size,
                              hipStream_t stream) {
    (void)in_sizes; (void)n_in; (void)out_size; (void)d_ws; (void)ws_size;
    const float* input   = (const float*)d_in[0];
    const float* phase   = (const float*)d_in[1];
    const float* weights = (const float*)d_in[2];
    const float* biases  = (const float*)d_in[3];
    float*       out     = (float*)d_out;

    dim3 grid(OUT_F / BN, BATCH / BM);   // 8 x 16 = 128 blocks
    phase_linear_wmma_kernel<<<grid, THREADS, 0, stream>>>(input, phase, weights, biases, out);
}